// LightGCN_67319317397757
// MI455X (gfx1250) — compile-verified
//
#include <hip/hip_runtime.h>

#define DIM   64
#define TILE  1024
#define BLOCK 256

typedef int v4i __attribute__((ext_vector_type(4)));

// ---------------------------------------------------------------------------
// CDNA5 helpers: async global->LDS copies (ASYNCcnt path) with asm fallback,
// relaxed agent-scope f32 atomic add (lowers to global/flat_atomic_add_f32).
// ---------------------------------------------------------------------------
__device__ __forceinline__ void async_copy_b32(const void* gsrc, void* ldst) {
#if __has_builtin(__builtin_amdgcn_global_load_async_to_lds_b32)
  __builtin_amdgcn_global_load_async_to_lds_b32(
      (__attribute__((address_space(1))) int*)gsrc,
      (__attribute__((address_space(3))) int*)ldst,
      /*imm offset*/ 0, /*cpol*/ 0);
#else
  unsigned lofs =
      (unsigned)(unsigned long long)(__attribute__((address_space(3))) void*)ldst;
  asm volatile("global_load_async_to_lds_b32 %0, %1, off"
               :
               : "v"(lofs), "v"(gsrc)
               : "memory");
#endif
}

__device__ __forceinline__ void async_copy_b128(const void* gsrc, void* ldst) {
#if __has_builtin(__builtin_amdgcn_global_load_async_to_lds_b128)
  __builtin_amdgcn_global_load_async_to_lds_b128(
      (__attribute__((address_space(1))) v4i*)gsrc,
      (__attribute__((address_space(3))) v4i*)ldst,
      /*imm offset*/ 0, /*cpol*/ 0);
#else
  unsigned lofs =
      (unsigned)(unsigned long long)(__attribute__((address_space(3))) void*)ldst;
  asm volatile("global_load_async_to_lds_b128 %0, %1, off"
               :
               : "v"(lofs), "v"(gsrc)
               : "memory");
#endif
}

__device__ __forceinline__ void wait_async0() {
#if __has_builtin(__builtin_amdgcn_s_wait_asynccnt)
  __builtin_amdgcn_s_wait_asynccnt(0);
#else
  asm volatile("s_wait_asynccnt 0" ::: "memory");
#endif
}

__device__ __forceinline__ void atomic_add_f32(float* p, float v) {
  __hip_atomic_fetch_add(p, v, __ATOMIC_RELAXED, __HIP_MEMORY_SCOPE_AGENT);
}

// ---------------------------------------------------------------------------
// 1) degree accumulation (weighted in-degree)
// ---------------------------------------------------------------------------
__global__ void lgcn_zero_kernel(float* __restrict__ p, int n) {
  int t = blockIdx.x * blockDim.x + threadIdx.x;
  if (t < n) p[t] = 0.0f;
}

__global__ void lgcn_deg_kernel(const int* __restrict__ dst,
                                const float* __restrict__ w,
                                float* __restrict__ deg, int E) {
  int e = blockIdx.x * blockDim.x + threadIdx.x;
  if (e < E) atomic_add_f32(&deg[dst[e]], w[e]);
}

__global__ void lgcn_norm_kernel(float* __restrict__ deg, int n) {
  int t = blockIdx.x * blockDim.x + threadIdx.x;
  if (t < n) {
    float d = deg[t];
    deg[t] = 1.0f / sqrtf(fmaxf(d, 1.0f));  // deg buffer becomes norm buffer
  }
}

// ---------------------------------------------------------------------------
// 2) node-wise elementwise kernels (float2-vectorized, total2 = N*DIM/2)
// ---------------------------------------------------------------------------
__global__ void lgcn_prep0_kernel(const float* __restrict__ h,
                                  const float* __restrict__ norm,
                                  float* __restrict__ hn, float* __restrict__ acc,
                                  float* __restrict__ out, int total2) {
  int t = blockIdx.x * blockDim.x + threadIdx.x;
  if (t >= total2) return;
  int node = t >> 5;  // DIM/2 = 32 float2 per row
  float nr = norm[node];
  float2 v = ((const float2*)h)[t];
  ((float2*)out)[t] = v;                              // emb sum starts at h0
  ((float2*)hn)[t]  = make_float2(v.x * nr, v.y * nr);
  ((float2*)acc)[t] = make_float2(0.0f, 0.0f);
}

__global__ void lgcn_post1_kernel(const float* __restrict__ norm,
                                  float* __restrict__ hn, float* __restrict__ acc,
                                  float* __restrict__ out, int total2) {
  int t = blockIdx.x * blockDim.x + threadIdx.x;
  if (t >= total2) return;
  int node = t >> 5;
  float nr = norm[node];
  float2 a = ((float2*)acc)[t];
  float2 h1 = make_float2(a.x * nr, a.y * nr);        // h1 = acc * norm
  float2 o = ((float2*)out)[t];
  ((float2*)out)[t] = make_float2(o.x + h1.x, o.y + h1.y);
  ((float2*)hn)[t]  = make_float2(h1.x * nr, h1.y * nr);
  ((float2*)acc)[t] = make_float2(0.0f, 0.0f);
}

__global__ void lgcn_post2_kernel(const float* __restrict__ norm,
                                  const float* __restrict__ acc,
                                  float* __restrict__ out, int total2) {
  int t = blockIdx.x * blockDim.x + threadIdx.x;
  if (t >= total2) return;
  int node = t >> 5;
  float nr = norm[node];
  const float inv3 = 1.0f / 3.0f;
  float2 a = ((const float2*)acc)[t];
  float2 o = ((float2*)out)[t];
  ((float2*)out)[t] = make_float2((o.x + a.x * nr) * inv3,
                                  (o.y + a.y * nr) * inv3);
}

// ---------------------------------------------------------------------------
// 3) edge pass: acc[dst] += w * hn[src]      (one 1024-edge tile per block)
//    - edge tuples staged into LDS with async b128 global->LDS loads
//    - half-wave per edge: 16 lanes x float4 covers the 256B row (b128 gather)
//    - scatter via native f32 global atomics (acc is L2-resident: 25.6 MB)
// ---------------------------------------------------------------------------
__global__ void __launch_bounds__(BLOCK) lgcn_edge_kernel(
    const float* __restrict__ hn, const float* __restrict__ w,
    const int* __restrict__ src, const int* __restrict__ dst,
    float* __restrict__ acc, int E) {
  __shared__ int   s_src[TILE];
  __shared__ int   s_dst[TILE];
  __shared__ float s_w[TILE];

  const int tid  = threadIdx.x;
  const int lane = tid & 31;
  const int wid  = tid >> 5;
  const int nwv  = BLOCK >> 5;
  const int base = blockIdx.x * TILE;
  const int rem  = E - base;
  const int n    = (rem < TILE) ? rem : TILE;

  if (n == TILE) {
    // Full tile: one async b128 per thread per array (16B/lane).
    async_copy_b128(src + base + tid * 4, &s_src[tid * 4]);
    async_copy_b128(dst + base + tid * 4, &s_dst[tid * 4]);
    async_copy_b128(w + base + tid * 4, &s_w[tid * 4]);
  } else {
    for (int i = tid; i < n; i += BLOCK) {
      async_copy_b32(src + base + i, &s_src[i]);
      async_copy_b32(dst + base + i, &s_dst[i]);
      async_copy_b32(w + base + i, &s_w[i]);
    }
  }
  wait_async0();       // each wave drains its own ASYNCcnt...
  __syncthreads();     // ...then all waves agree the LDS tile is valid

  // Two edges per wave: lanes 0-15 -> edge 2p, lanes 16-31 -> edge 2p+1.
  const int half   = lane >> 4;
  const int l16    = lane & 15;
  const int npairs = (n + 1) >> 1;
  for (int p = wid; p < npairs; p += nwv) {
    int e = 2 * p + half;
    if (e < n) {
      // Prefetch the row this half-wave gathers next iteration.
      int en = e + 2 * nwv;
      if (en < n)
        __builtin_prefetch(hn + (size_t)s_src[en] * DIM + l16 * 4, 0, 0);

      int   sn = s_src[e];
      int   dn = s_dst[e];
      float ww = s_w[e];
      float4 v = *(const float4*)(hn + (size_t)sn * DIM + l16 * 4);
      float* ap = acc + (size_t)dn * DIM + l16 * 4;
      atomic_add_f32(ap + 0, v.x * ww);
      atomic_add_f32(ap + 1, v.y * ww);
      atomic_add_f32(ap + 2, v.z * ww);
      atomic_add_f32(ap + 3, v.w * ww);
    }
  }
}

// ---------------------------------------------------------------------------
// Launch: zero -> deg -> norm -> prep0 -> edge -> post1 -> edge -> post2
// ---------------------------------------------------------------------------
extern "C" void kernel_launch(void* const* d_in, const int* in_sizes, int n_in,
                              void* d_out, int out_size, void* d_ws, size_t ws_size,
                              hipStream_t stream) {
  const float* h   = (const float*)d_in[0];
  const float* w   = (const float*)d_in[1];
  const int*   src = (const int*)d_in[2];
  const int*   dst = (const int*)d_in[3];
  const int N = in_sizes[0] / DIM;
  const int E = in_sizes[1];
  float* out = (float*)d_out;

  // Workspace layout: norm [N] | hn [N*DIM] | acc [N*DIM]  (~52 MB)
  float* norm = (float*)d_ws;
  float* hn   = norm + N;
  float* acc  = hn + (size_t)N * DIM;

  const int gN     = (N + BLOCK - 1) / BLOCK;
  const int gE     = (E + BLOCK - 1) / BLOCK;
  const int total2 = N * (DIM / 2);
  const int g2     = (total2 + BLOCK - 1) / BLOCK;
  const int tiles  = (E + TILE - 1) / TILE;

  lgcn_zero_kernel<<<gN, BLOCK, 0, stream>>>(norm, N);
  lgcn_deg_kernel<<<gE, BLOCK, 0, stream>>>(dst, w, norm, E);
  lgcn_norm_kernel<<<gN, BLOCK, 0, stream>>>(norm, N);

  lgcn_prep0_kernel<<<g2, BLOCK, 0, stream>>>(h, norm, hn, acc, out, total2);
  lgcn_edge_kernel<<<tiles, BLOCK, 0, stream>>>(hn, w, src, dst, acc, E);
  lgcn_post1_kernel<<<g2, BLOCK, 0, stream>>>(norm, hn, acc, out, total2);
  lgcn_edge_kernel<<<tiles, BLOCK, 0, stream>>>(hn, w, src, dst, acc, E);
  lgcn_post2_kernel<<<g2, BLOCK, 0, stream>>>(norm, acc, out, total2);
}